// RendererModel_43602507989576
// MI455X (gfx1250) — compile-verified
//
#include <hip/hip_runtime.h>
#include <hip/hip_bf16.h>

typedef float v2f __attribute__((ext_vector_type(2)));
typedef float v4f __attribute__((ext_vector_type(4)));
typedef float v8f __attribute__((ext_vector_type(8)));

#define S_DIM   256
#define T_DIM   4
#define INF_F   1000000000.0f
#define DEG2RAD 0.017453292519943295f

__device__ __forceinline__ void rot_vert(const float* __restrict__ v, int idx,
                                         float ca, float sa, float ce, float se,
                                         float& X, float& Y, float& Z) {
  float x = v[idx*3+0], y = v[idx*3+1], z = v[idx*3+2];
  // R = Rx(ele) @ Ry(azi)
  X = ca*x + sa*z;
  Y = se*sa*x + ce*y - se*ca*z;
  Z = -ce*sa*x + se*y + ce*ca*z;
}

// Per-face affine coefficients for (w0, w1, w2, depth), pre-swizzled into the
// WMMA 16x4 f32 A-operand lane layout and packed 8 floats per (group, lane):
//   lane m (0..15):  { A0,B0 | A1,B1 | A2,B2 | P,Q  }   (K=0,1 rows)
//   lane m+16:       { C0,0  | C1,0  | C2,0  | Rc,0 }   (K=2,3 rows)
__global__ void prep_faces(const float* __restrict__ verts,
                           const int*   __restrict__ faces,
                           const int*   __restrict__ azi_p,
                           const int*   __restrict__ ele_p,
                           v4f* __restrict__ coef,
                           int F0, int ngroups)
{
  int t = blockIdx.x * blockDim.x + threadIdx.x;
  int g = t >> 5;
  int lane = t & 31;
  if (g >= ngroups) return;
  int m = lane & 15;
  int f = g * 16 + m;
  bool hi = (lane >= 16);

  float A0 = 0.f, B0 = 0.f, C0 = 0.f;
  float A1 = 0.f, B1 = 0.f, C1 = -1e30f;   // default: face excluded (w1 < 0)
  float A2 = 0.f, B2 = 0.f, C2 = 0.f;
  float P = 0.f, Q = 0.f, Rc = 0.f;

  if (f < F0) {
    float a = (float)azi_p[0] * DEG2RAD;
    float e = (float)ele_p[0] * DEG2RAD;
    float ca = __cosf(a), sa = __sinf(a), ce = __cosf(e), se = __sinf(e);
    int i0 = faces[f*3+0], i1 = faces[f*3+1], i2 = faces[f*3+2];
    float x0,y0,z0, x1,y1,z1, x2,y2,z2;
    rot_vert(verts, i0, ca, sa, ce, se, x0, y0, z0);
    rot_vert(verts, i1, ca, sa, ce, se, x1, y1, z1);
    rot_vert(verts, i2, ca, sa, ce, se, x2, y2, z2);
    float e1x = x1-x0, e1y = y1-y0, e2x = x2-x0, e2y = y2-y0;
    float det = e1x*e2y - e1y*e2x;
    bool valid = fabsf(det) >= 1e-8f;
    float inv = 1.0f / (valid ? det : 1.0f);
    // w1 = px*A1 + py*B1 + C1 ; w2 = px*A2 + py*B2 + C2 ; w0 = 1 - w1 - w2
    A1 =  e2y * inv;  B1 = -e2x * inv;
    float C1r = (y0*e2x - x0*e2y) * inv;
    A2 = -e1y * inv;  B2 =  e1x * inv;
    float C2r = (x0*e1y - y0*e1x) * inv;
    A0 = -(A1 + A2);  B0 = -(B1 + B2);  C0 = 1.0f - C1r - C2r;
    // depth = z0 + w1*(z1-z0) + w2*(z2-z0)  -> affine in (px,py)
    float dz1 = z1 - z0, dz2 = z2 - z0;
    P  = A1*dz1 + A2*dz2;
    Q  = B1*dz1 + B2*dz2;
    Rc = z0 + C1r*dz1 + C2r*dz2;
    C1 = valid ? C1r : -1e30f;   // degenerate faces never pass inside-test
    C2 = C2r;
  }

  v4f q0, q1;
  if (!hi) { q0.x = A0; q0.y = B0; q0.z = A1; q0.w = B1;
             q1.x = A2; q1.y = B2; q1.z = P;  q1.w = Q;  }
  else     { q0.x = C0; q0.y = 0.f; q0.z = C1; q0.w = 0.f;
             q1.x = C2; q1.y = 0.f; q1.z = Rc; q1.w = 0.f; }
  coef[(g*32 + lane)*2 + 0] = q0;
  coef[(g*32 + lane)*2 + 1] = q1;
}

// One wave per 16 pixels. Faces on M (A operand), pixels on N (B operand).
// C layout: lanes 0-15 hold faces m=0..7 (VGPR r), lanes 16-31 hold m=8..15.
__global__ __launch_bounds__(256) void raster_wmma(
    const float* __restrict__ verts,
    const int*   __restrict__ faces,
    const float* __restrict__ teximg,
    const float* __restrict__ tgrid,
    const int*   __restrict__ azi_p,
    const int*   __restrict__ ele_p,
    const v4f*   __restrict__ coef,
    float* __restrict__ out,
    int F0, int ngroups)
{
  const int lane = threadIdx.x & 31;
  const int wave = blockIdx.x * (blockDim.x >> 5) + (threadIdx.x >> 5);
  const int n  = lane & 15;
  const int p  = wave * 16 + n;         // flat pixel id
  const int pi = p >> 8;                // row
  const int pj = p & (S_DIM - 1);       // col
  const float px = ((float)pj + 0.5f) * (2.0f / (float)S_DIM) - 1.0f;
  const float py = -(((float)pi + 0.5f) * (2.0f / (float)S_DIM) - 1.0f);

  // B operand (4x16 f32): lane n holds K=0,1 = {px,py}; lane n+16 holds K=2,3 = {1,0}
  v2f bvec;
  bvec.x = (lane < 16) ? px : 1.0f;
  bvec.y = (lane < 16) ? py : 0.0f;

  float bestD = INF_F;
  int   bestF = 0;
  const int baseM = (lane < 16) ? 0 : 8;

#pragma unroll 2
  for (int g = 0; g < ngroups; ++g) {
    v4f q0 = coef[(g*32 + lane)*2 + 0];
    v4f q1 = coef[(g*32 + lane)*2 + 1];
    v2f a0; a0.x = q0.x; a0.y = q0.y;
    v2f a1; a1.x = q0.z; a1.y = q0.w;
    v2f a2; a2.x = q1.x; a2.y = q1.y;
    v2f a3; a3.x = q1.z; a3.y = q1.w;
    v8f zc = {};
    v8f w0 = __builtin_amdgcn_wmma_f32_16x16x4_f32(false, a0, false, bvec, (short)0, zc, false, false);
    v8f w1 = __builtin_amdgcn_wmma_f32_16x16x4_f32(false, a1, false, bvec, (short)0, zc, false, false);
    v8f w2 = __builtin_amdgcn_wmma_f32_16x16x4_f32(false, a2, false, bvec, (short)0, zc, false, false);
    v8f dp = __builtin_amdgcn_wmma_f32_16x16x4_f32(false, a3, false, bvec, (short)0, zc, false, false);
#pragma unroll
    for (int r = 0; r < 8; ++r) {
      float mn = fminf(fminf(w0[r], w1[r]), w2[r]);   // inside <=> mn >= 0
      float d  = dp[r];
      bool c = (mn >= 0.0f) && (d < bestD);           // strict <: earliest face id
      bestF = c ? (g*16 + baseM + r) : bestF;
      bestD = c ? d : bestD;
    }
  }

  // Combine the two lane-halves (faces m=0..7 vs m=8..15); tie -> smaller face id
  float dO = __shfl_xor(bestD, 16, 32);
  int   fO = __shfl_xor(bestF, 16, 32);
  bool take = (dO < bestD) || ((dO == bestD) && (fO < bestF));
  bestD = take ? dO : bestD;
  bestF = take ? fO : bestF;

  if (lane < 16) {
    float c0 = 0.f, c1 = 0.f, c2 = 0.f;
    if (bestD < INF_F) {
      int f = bestF;
      float a = (float)azi_p[0] * DEG2RAD;
      float e = (float)ele_p[0] * DEG2RAD;
      float ca = __cosf(a), sa = __sinf(a), ce = __cosf(e), se = __sinf(e);
      int i0 = faces[f*3+0], i1 = faces[f*3+1], i2 = faces[f*3+2];
      float x0,y0,z0, x1,y1,z1, x2,y2,z2;
      rot_vert(verts, i0, ca, sa, ce, se, x0, y0, z0);
      rot_vert(verts, i1, ca, sa, ce, se, x1, y1, z1);
      rot_vert(verts, i2, ca, sa, ce, se, x2, y2, z2);
      float e1x = x1-x0, e1y = y1-y0, e2x = x2-x0, e2y = y2-y0;
      float det = e1x*e2y - e1y*e2x;
      float dets = (fabsf(det) < 1e-8f) ? 1.0f : det;
      float dx = px - x0, dy = py - y0;
      float W1 = (dx*e2y - dy*e2x) / dets;
      float W2 = (e1x*dy - e1y*dx) / dets;
      float W0 = 1.0f - W1 - W2;
      int t0 = min(max((int)floorf(W0 * (float)T_DIM), 0), T_DIM - 1);
      int t1 = min(max((int)floorf(W1 * (float)T_DIM), 0), T_DIM - 1);
      int t2 = min(max((int)floorf(W2 * (float)T_DIM), 0), T_DIM - 1);
      int k = (t0 * T_DIM + t1) * T_DIM + t2;     // 0..63
      float gx = tgrid[(f*64 + k)*2 + 0];
      float gy = tgrid[(f*64 + k)*2 + 1];
      // bilinear sample of 1024x1024, 3 channels
      float X = (gx + 1.0f) * 512.0f - 0.5f;
      float Y = (gy + 1.0f) * 512.0f - 0.5f;
      float xf = floorf(X), yf = floorf(Y);
      float wx = X - xf, wy = Y - yf;
      int xi = (int)xf, yi = (int)yf;
      float w00 = (1.f-wx)*(1.f-wy), w10 = wx*(1.f-wy);
      float w01 = (1.f-wx)*wy,       w11 = wx*wy;
      float acc[3];
#pragma unroll
      for (int c = 0; c < 3; ++c) {
        const float* img = teximg + (size_t)c * 1024 * 1024;
        auto tap = [&](int xx, int yy, float wgt) -> float {
          bool valid = (xx >= 0) && (xx < 1024) && (yy >= 0) && (yy < 1024);
          int xc = min(max(xx, 0), 1023);
          int yc = min(max(yy, 0), 1023);
          return img[yc * 1024 + xc] * (valid ? wgt : 0.0f);
        };
        acc[c] = tap(xi,   yi,   w00) + tap(xi+1, yi,   w10)
               + tap(xi,   yi+1, w01) + tap(xi+1, yi+1, w11);
      }
      c0 = acc[0]; c1 = acc[1]; c2 = acc[2];
    }
    // out[0, c, i, j] = color[i, S-1-j, c]   (transpose(2,0,1) then W-reversal)
    int jout = (S_DIM - 1) - pj;
    out[0 * S_DIM * S_DIM + pi * S_DIM + jout] = c0;
    out[1 * S_DIM * S_DIM + pi * S_DIM + jout] = c1;
    out[2 * S_DIM * S_DIM + pi * S_DIM + jout] = c2;
  }
}

extern "C" void kernel_launch(void* const* d_in, const int* in_sizes, int n_in,
                              void* d_out, int out_size, void* d_ws, size_t ws_size,
                              hipStream_t stream) {
  const float* verts  = (const float*)d_in[0];
  const float* teximg = (const float*)d_in[1];
  const float* tgrid  = (const float*)d_in[2];
  const int*   faces  = (const int*)d_in[3];
  const int*   azi    = (const int*)d_in[4];
  const int*   ele    = (const int*)d_in[5];
  float* out = (float*)d_out;

  int F0 = in_sizes[3] / 3;              // 6016
  int ngroups = (F0 + 15) / 16;          // 376

  v4f* coef = (v4f*)d_ws;                // ngroups*32*2 v4f  (~385 KB)

  int prepThreads = ngroups * 32;
  prep_faces<<<(prepThreads + 255) / 256, 256, 0, stream>>>(
      verts, faces, azi, ele, coef, F0, ngroups);

  int waves  = (S_DIM * S_DIM) / 16;     // 4096 waves, one per 16-pixel group
  int blocks = waves / 8;                // 512 blocks x 256 threads (8 waves)
  raster_wmma<<<blocks, 256, 0, stream>>>(
      verts, faces, teximg, tgrid, azi, ele, coef, out, F0, ngroups);
}